// HintGenKernel_8057358647761
// MI455X (gfx1250) — compile-verified
//
#include <hip/hip_runtime.h>
#include <stdint.h>

// Gather + variable-length segment XOR reduction for MI455X (gfx1250).
//
// One hint (segment) per wave32. Each wave:
//   1. async-stages its (block, offset) index pairs interleaved into LDS via
//      gfx1250 GLOBAL_LOAD_ASYNC_TO_LDS_B64 with th:TH_LOAD_NT (read-once
//      stream must not evict the L2-resident 40MB entries table),
//   2. s_wait_asynccnt 0, then lane-strided gather of 40B rows (default RT
//      temporal hint -> stays in L2) with one-iteration-ahead prefetch,
//   3. butterfly XOR reduction across the wave with __shfl_xor (wave32: 5 steps),
//   4. lane 0 stores the 5 int64 parities non-temporally (write-once output).

#define WAVES_PER_BLOCK 8
#define MAX_SEG         128   // MAXSZ = 128 -> per-hint size <= 127
#define THREADS         (WAVES_PER_BLOCK * 32)

__global__ __launch_bounds__(THREADS, 4)
void seg_xor_gather_kernel(const long long* __restrict__ entries,
                           const long long* __restrict__ sub_blocks,
                           const long long* __restrict__ sub_offsets,
                           const long long* __restrict__ sub_starts,
                           const long long* __restrict__ sub_sizes,
                           const int*       __restrict__ block_size_ptr,
                           unsigned long long* __restrict__ out,
                           int H, long long N)
{
    // Interleaved (block, offset) pairs: element t lives at byte offset t*16,
    // always 16B-aligned -> phase 2 reads one ds_load_b128 per element.
    __shared__ __align__(16) unsigned long long s_pair[WAVES_PER_BLOCK][MAX_SEG][2];

    const int lane = threadIdx.x & 31;
    const int wave = threadIdx.x >> 5;
    const int h    = blockIdx.x * WAVES_PER_BLOCK + wave;
    if (h >= H) return;   // whole wave exits together

    const long long start = sub_starts[h];
    const int       size  = (int)sub_sizes[h];
    const long long BS    = (long long)block_size_ptr[0];  // low 32 bits: OK for i32 or i64 LE

    // ---- Phase 1: async-stage the ragged index slice into LDS (gfx1250 path).
    // Generic pointers to LDS carry the LDS byte offset in their low 32 bits.
    // NT hint: this stream is consumed exactly once; keep L2 for `entries`.
    for (int t = lane; t < size; t += 32) {
        unsigned lb = (unsigned)(uintptr_t)&s_pair[wave][t][0];
        unsigned lo = (unsigned)(uintptr_t)&s_pair[wave][t][1];
        unsigned long long gb = (unsigned long long)(uintptr_t)&sub_blocks[start + t];
        unsigned long long go = (unsigned long long)(uintptr_t)&sub_offsets[start + t];
        asm volatile("global_load_async_to_lds_b64 %0, %2, off th:TH_LOAD_NT\n\t"
                     "global_load_async_to_lds_b64 %1, %3, off th:TH_LOAD_NT"
                     :: "v"(lb), "v"(lo), "v"(gb), "v"(go)
                     : "memory");
    }
    // Wave consumes only its own staged region -> waiting on ASYNCcnt is enough.
    asm volatile("s_wait_asynccnt 0x0" ::: "memory");

    // ---- Phase 2: lane-strided gather + local XOR accumulation.
    unsigned long long a0 = 0, a1 = 0, a2 = 0, a3 = 0, a4 = 0;
    for (int t = lane; t < size; t += 32) {
        const ulonglong2 bo = *(const ulonglong2*)&s_pair[wave][t][0];  // one b128 DS read
        long long idx = (long long)bo.x * BS + (long long)bo.y;
        idx = (idx < 0) ? 0 : ((idx >= N) ? (N - 1) : idx);             // reference's clip

        // Prefetch next iteration's row to hide L2 latency on the gather chain.
        const int tn = t + 32;
        if (tn < size) {
            const ulonglong2 bon = *(const ulonglong2*)&s_pair[wave][tn][0];
            long long idxn = (long long)bon.x * BS + (long long)bon.y;
            idxn = (idxn < 0) ? 0 : ((idxn >= N) ? (N - 1) : idxn);
            __builtin_prefetch((const void*)(entries + idxn * 5), 0, 0);
        }

        const unsigned long long* row = (const unsigned long long*)(entries + idx * 5);
        a0 ^= row[0];
        a1 ^= row[1];
        a2 ^= row[2];
        a3 ^= row[3];
        a4 ^= row[4];
    }

    // ---- Phase 3: butterfly XOR reduction across the wave (wave32: 5 steps).
    for (int m = 16; m >= 1; m >>= 1) {
        a0 ^= __shfl_xor(a0, m, 32);
        a1 ^= __shfl_xor(a1, m, 32);
        a2 ^= __shfl_xor(a2, m, 32);
        a3 ^= __shfl_xor(a3, m, 32);
        a4 ^= __shfl_xor(a4, m, 32);
    }

    // ---- Phase 4: lane 0 writes the 5 parities non-temporally (write-once).
    if (lane == 0) {
        unsigned long long* o = out + (size_t)h * 5;
        __builtin_nontemporal_store(a0, &o[0]);
        __builtin_nontemporal_store(a1, &o[1]);
        __builtin_nontemporal_store(a2, &o[2]);
        __builtin_nontemporal_store(a3, &o[3]);
        __builtin_nontemporal_store(a4, &o[4]);
    }
}

extern "C" void kernel_launch(void* const* d_in, const int* in_sizes, int n_in,
                              void* d_out, int out_size, void* d_ws, size_t ws_size,
                              hipStream_t stream)
{
    const long long* entries     = (const long long*)d_in[0];
    const long long* sub_blocks  = (const long long*)d_in[1];
    const long long* sub_offsets = (const long long*)d_in[2];
    const long long* sub_starts  = (const long long*)d_in[3];
    const long long* sub_sizes   = (const long long*)d_in[4];
    const int*       bs_ptr      = (const int*)d_in[5];
    unsigned long long* out      = (unsigned long long*)d_out;

    const int       H = in_sizes[3];                 // number of hints
    const long long N = (long long)in_sizes[0] / 5;  // rows of `entries`

    const int grid = (H + WAVES_PER_BLOCK - 1) / WAVES_PER_BLOCK;
    hipLaunchKernelGGL(seg_xor_gather_kernel, dim3(grid), dim3(THREADS), 0, stream,
                       entries, sub_blocks, sub_offsets, sub_starts, sub_sizes,
                       bs_ptr, out, H, N);
}